// Marginal_CrossEntropy_41575283425498
// MI455X (gfx1250) — compile-verified
//
#include <hip/hip_runtime.h>

typedef __attribute__((ext_vector_type(2))) float v2f;
typedef __attribute__((ext_vector_type(8))) float v8f;

#define SMOOTH_F 1e-5f
#define C_CH     24
#define SPATIAL  884736    /* 96*96*96            */
#define SPATIAL4 221184    /* SPATIAL / 4         */
#define NVOX     1769472   /* 2 * SPATIAL         */
#define NITEMS   442368    /* NVOX / 4            */
#define BLOCK    256
#define NBLOCKS  1728      /* NITEMS / BLOCK (exact) */

// Wave32 sum via the matrix pipe: A = partials (lanes 0-15 -> A[m,0],
// lanes 16-31 -> A[m,2], other K rows zero), B = ones. D[m,n] = p_m + p_{m+16}.
// Lane n (n<16) holds D[0..7, n] in the 8 acc VGPRs, lane n+16 holds D[8..15, n].
// Sum the 8 regs, then fold lane n <-> n+16. EXEC must be all-1s at the call.
__device__ __forceinline__ float wave_sum_wmma(float p) {
  v2f a; a[0] = p;    a[1] = 0.0f;
  v2f b; b[0] = 1.0f; b[1] = 1.0f;
  v8f acc = {};
  acc = __builtin_amdgcn_wmma_f32_16x16x4_f32(
      /*neg_a=*/false, a, /*neg_b=*/false, b,
      /*c_mod=*/(short)0, acc, /*reuse_a=*/false, /*reuse_b=*/false);
  float s = acc[0] + acc[1] + acc[2] + acc[3] + acc[4] + acc[5] + acc[6] + acc[7];
  s += __shfl_xor(s, 16, 32);
  return s;
}

__device__ __forceinline__ float voxel_contrib(int t, float g, float asum, unsigned m) {
  // t==0: value = x0 + absent_sum (mask[0] is always 1); else value = x_t.
  float v  = (t == 0) ? (g + asum) : g;
  float lp = __logf(fminf(fmaxf(v, SMOOTH_F), 1.0f));
  // alpha = mask[t]; accumulate +(logpt + SMOOTH); final kernel negates.
  return ((m >> t) & 1u) ? (lp + SMOOTH_F) : 0.0f;
}

__global__ void __launch_bounds__(BLOCK)
mce_partial_kernel(const float* __restrict__ logit0,
                   const int*   __restrict__ target,
                   const int*   __restrict__ class_for_batch,
                   float*       __restrict__ partial) {
  __shared__ unsigned s_mask;
  __shared__ float    s_wave[BLOCK / 32];

  if (threadIdx.x == 0) {
    unsigned m = 1u;                       // class 0 always present
    for (int i = 0; i < 12; ++i) m |= 1u << (class_for_batch[i] & 31);
    s_mask = m;
  }
  __syncthreads();
  const unsigned m = s_mask;

  const int i  = blockIdx.x * BLOCK + threadIdx.x;   // 0 .. NITEMS-1, exact
  const int b  = i / SPATIAL4;                        // batch (0 or 1)
  const int s4 = i - b * SPATIAL4;                    // float4 index in spatial

  const float4* base = (const float4*)logit0 + (size_t)b * (C_CH * SPATIAL4) + s4;
  const int4 tgt = ((const int4*)target)[i];          // target is [B,1,D,H,W]

  float g0 = 0.f, g1 = 0.f, g2 = 0.f, g3 = 0.f;       // gathered x[tgt]
  float a0 = 0.f, a1 = 0.f, a2 = 0.f, a3 = 0.f;       // absent-class sum

#pragma unroll
  for (int c = 0; c < C_CH; ++c) {
    float4 x = base[c * SPATIAL4];                    // coalesced b128 stream
    if (tgt.x == c) g0 = x.x;
    if (tgt.y == c) g1 = x.y;
    if (tgt.z == c) g2 = x.z;
    if (tgt.w == c) g3 = x.w;
    if (!((m >> c) & 1u)) { a0 += x.x; a1 += x.y; a2 += x.z; a3 += x.w; }
  }

  float p = voxel_contrib(tgt.x, g0, a0, m)
          + voxel_contrib(tgt.y, g1, a1, m)
          + voxel_contrib(tgt.z, g2, a2, m)
          + voxel_contrib(tgt.w, g3, a3, m);

  // Wave reduction on the matrix pipe (full EXEC here: no tail, no divergence).
  float w = wave_sum_wmma(p);
  const int lane = threadIdx.x & 31;
  const int wid  = threadIdx.x >> 5;
  if (lane == 0) s_wave[wid] = w;
  __syncthreads();
  if (threadIdx.x == 0) {
    float t = 0.f;
#pragma unroll
    for (int j = 0; j < BLOCK / 32; ++j) t += s_wave[j];
    partial[blockIdx.x] = t;                          // deterministic: no atomics
  }
}

__global__ void __launch_bounds__(BLOCK)
mce_final_kernel(const float* __restrict__ partial, float* __restrict__ out) {
  __shared__ float s_wave[BLOCK / 32];

  float p = 0.f;
  for (int i = threadIdx.x; i < NBLOCKS; i += BLOCK) p += partial[i];  // fixed order

  float w = wave_sum_wmma(p);   // all 256 lanes reconverged here
  const int lane = threadIdx.x & 31;
  const int wid  = threadIdx.x >> 5;
  if (lane == 0) s_wave[wid] = w;
  __syncthreads();
  if (threadIdx.x == 0) {
    float t = 0.f;
#pragma unroll
    for (int j = 0; j < BLOCK / 32; ++j) t += s_wave[j];
    out[0] = -t / (float)NVOX;   // loss = mean(-alpha*ce)
  }
}

extern "C" void kernel_launch(void* const* d_in, const int* in_sizes, int n_in,
                              void* d_out, int out_size, void* d_ws, size_t ws_size,
                              hipStream_t stream) {
  const float* logit0          = (const float*)d_in[0];
  const int*   target          = (const int*)d_in[1];
  const int*   class_for_batch = (const int*)d_in[2];
  float*       partial         = (float*)d_ws;        // NBLOCKS floats

  mce_partial_kernel<<<NBLOCKS, BLOCK, 0, stream>>>(logit0, target, class_for_batch, partial);
  mce_final_kernel<<<1, BLOCK, 0, stream>>>(partial, (float*)d_out);
}